// Embedding_31147102831089
// MI455X (gfx1250) — compile-verified
//
#include <hip/hip_runtime.h>
#include <hip/hip_bf16.h>
#include <stdint.h>

// ---------------------------------------------------------------------------
// Embedding gather for MI455X (gfx1250): pure bandwidth problem (~135 MB of
// traffic => ~6 us at 23.3 TB/s). No FLOPs => no WMMA. CDNA5-specific path:
// async global<->LDS B128 copies (ASYNCcnt) when the builtins exist, else
// float4 (global_load_b128/global_store_b128) + global_prefetch_b8.
// ---------------------------------------------------------------------------

#define B_DIM 32
#define S_DIM 32
#define W_DIM 30
#define E_DIM 300
#define CHUNKS 75        // 300 floats = 75 x float4 (16B); rows are 1200B, 16B aligned
#define WAVES_PER_BLOCK 8
#define LDS_ROW_F 320    // padded row size in floats per staging buffer

#if defined(__HIP_DEVICE_COMPILE__) && defined(__gfx1250__) && \
    __has_builtin(__builtin_amdgcn_global_load_async_to_lds_b128) && \
    __has_builtin(__builtin_amdgcn_global_store_async_from_lds_b128)
#define USE_ASYNC_LDS 1
#else
#define USE_ASYNC_LDS 0
#endif

#if USE_ASYNC_LDS
// The builtin signature (from clang's diagnostic) is:
//   (AS1 v4i* global, AS3 v4i* lds, imm int offset, imm int cpol)
// where v4i is the GCC-style 16-byte int vector.
typedef int v4i __attribute__((vector_size(4 * sizeof(int))));
typedef __attribute__((address_space(1))) v4i* g_v4i_p;   // prints as __device__ v4i*
typedef __attribute__((address_space(3))) v4i* l_v4i_p;   // LDS

__device__ __forceinline__ void wait_asynccnt0() {
#if __has_builtin(__builtin_amdgcn_s_wait_asynccnt)
  __builtin_amdgcn_s_wait_asynccnt(0);
#else
  asm volatile("s_wait_asynccnt 0" ::: "memory");
#endif
}
#endif

// ---------------------------------------------------------------------------
// Kernel 1: per-sentence mask/temp computation. One wave32 per sentence.
// Reference's inner break-on-first-zero loop == keep tokens before the first
// zero; done with a single wave ballot.
// ---------------------------------------------------------------------------
__global__ __launch_bounds__(32) void mask_kernel(
    const int* __restrict__ idx,      // [1024*30]
    int* __restrict__ temp_ws,        // [1024*30]
    float* __restrict__ mask,         // [32, 960] -> flat == s_lin*30+w
    float* __restrict__ word_mask,    // [1024, 30]
    float* __restrict__ sent_mask)    // [32, 32]  -> flat == s_lin
{
  const int s_lin = blockIdx.x;              // 0..1023 (= b*32 + s)
  const int w     = threadIdx.x;             // 0..31 (wave32)
  const int base  = s_lin * W_DIM;

  int v = 0;
  if (w < W_DIM) v = idx[base + w];
  const bool nz = (w < W_DIM) && (v != 0);

  const unsigned long long m = __ballot(nz);     // bits 0..29 meaningful
  // first zero position (lane 30 bit of ~m is always set -> caps at 30)
  const int first_zero = __ffsll(~m) - 1;
  const bool keep = (w < first_zero);
  const int t = keep ? v : 0;

  if (w < W_DIM) {
    temp_ws[base + w]   = t;
    word_mask[base + w] = nz ? 1.0f : 0.0f;
    mask[base + w]      = (t != 0) ? 1.0f : 0.0f;
  }
  if (w == 0) sent_mask[s_lin] = (m != 0ull) ? 1.0f : 0.0f;
}

// ---------------------------------------------------------------------------
// Kernel 2: one wave per token. Stage the 1200B embedding row(s) in LDS via
// async B128 copies, then async-store to both destinations (reusing the
// staged row when temp == idx, the common case).
// ---------------------------------------------------------------------------
__global__ __launch_bounds__(WAVES_PER_BLOCK * 32) void gather_kernel(
    const int* __restrict__ idx,
    const int* __restrict__ temp_ws,
    const float* __restrict__ W,      // [50000, 300]
    float* __restrict__ emb,          // [30720, 300]
    float* __restrict__ share,        // [30720, 300]
    int ntok)
{
#if USE_ASYNC_LDS
  __shared__ __align__(16) float lds[WAVES_PER_BLOCK * 2 * LDS_ROW_F];
#endif
  const int lane    = threadIdx.x & 31;
  const int wave    = threadIdx.x >> 5;
  const int wave_id = blockIdx.x * WAVES_PER_BLOCK + wave;
  const int nwaves  = gridDim.x * WAVES_PER_BLOCK;

#if USE_ASYNC_LDS
  float* bufA = &lds[wave * 2 * LDS_ROW_F];
  float* bufB = bufA + LDS_ROW_F;
#endif

  for (int t = wave_id; t < ntok; t += nwaves) {
    const int row1 = idx[t];       // wave-uniform
    const int row2 = temp_ws[t];   // wave-uniform
    const bool same = (row2 == row1);

    const float* src1 = W + (size_t)row1 * E_DIM;
    const float* src2 = W + (size_t)row2 * E_DIM;
    float* d1 = emb   + (size_t)t * E_DIM;
    float* d2 = share + (size_t)t * E_DIM;

#if USE_ASYNC_LDS
    // Stage rows into LDS: 75 x b128 per row, 3 passes of 32 lanes.
    for (int c = lane; c < CHUNKS; c += 32) {
      __builtin_amdgcn_global_load_async_to_lds_b128(
          (g_v4i_p)(src1 + c * 4), (l_v4i_p)(bufA + c * 4), 0, 0);
    }
    if (!same) {
      for (int c = lane; c < CHUNKS; c += 32) {
        __builtin_amdgcn_global_load_async_to_lds_b128(
            (g_v4i_p)(src2 + c * 4), (l_v4i_p)(bufB + c * 4), 0, 0);
      }
    }
    wait_asynccnt0();  // LDS staging complete before stores read it

    float* srcB = same ? bufA : bufB;
    for (int c = lane; c < CHUNKS; c += 32) {
      __builtin_amdgcn_global_store_async_from_lds_b128(
          (g_v4i_p)(d1 + c * 4), (l_v4i_p)(bufA + c * 4), 0, 0);
      __builtin_amdgcn_global_store_async_from_lds_b128(
          (g_v4i_p)(d2 + c * 4), (l_v4i_p)(srcB + c * 4), 0, 0);
    }
    wait_asynccnt0();  // stores done reading LDS before next-iter overwrite
#else
    // Fallback: 128-bit VGPR path + speculative prefetch of next token's row.
    const int tn = t + nwaves;
    if (tn < ntok) {
      __builtin_prefetch(W + (size_t)idx[tn] * E_DIM, 0, 0);
    }
    const float4* s1 = (const float4*)src1;
    const float4* s2 = (const float4*)src2;
    float4* o1 = (float4*)d1;
    float4* o2 = (float4*)d2;
    for (int c = lane; c < CHUNKS; c += 32) {
      const float4 v1 = s1[c];
      o1[c] = v1;
      o2[c] = same ? v1 : s2[c];
    }
#endif
  }
}

extern "C" void kernel_launch(void* const* d_in, const int* in_sizes, int n_in,
                              void* d_out, int out_size, void* d_ws, size_t ws_size,
                              hipStream_t stream) {
  const int*   idx = (const int*)d_in[0];   // input_var [32,32,30] (int)
  const float* W   = (const float*)d_in[1]; // W_emb [50000,300] (f32)

  const int ntok = B_DIM * S_DIM * W_DIM;   // 30720

  float* out       = (float*)d_out;
  float* emb       = out;                                    //  9,216,000
  float* share     = emb + (size_t)ntok * E_DIM;             //  9,216,000
  float* mask      = share + (size_t)ntok * E_DIM;           //     30,720
  float* word_mask = mask + ntok;                            //     30,720
  float* sent_mask = word_mask + ntok;                       //      1,024

  int* temp_ws = (int*)d_ws;                // 30720 ints of scratch

  mask_kernel<<<B_DIM * S_DIM, 32, 0, stream>>>(
      idx, temp_ws, mask, word_mask, sent_mask);

  // 1920 blocks * 8 waves = 15360 waves -> 2 tokens per wave.
  gather_kernel<<<1920, WAVES_PER_BLOCK * 32, 0, stream>>>(
      idx, temp_ws, W, emb, share, ntok);
}